// SuperResolve_14989435863551
// MI455X (gfx1250) — compile-verified
//
#include <hip/hip_runtime.h>

// Fused 5-wavelet periodic DWT -> softshrink-delta -> iDWT -> weighted sum.
// All filtering stages are banded matmuls on V_WMMA_F32_16X16X4_F32 (wave32).
// 64x64 output tile per workgroup (512 thr = 16 waves); 104x112 halo in LDS.
// Band-matrix fragments are prebuilt per wavelet into LDS tables so every
// WMMA operand is one aligned ds_load_b64; fragments for a whole tile-job are
// preloaded into registers before the (fully unrolled) wmma chain issues.

typedef __attribute__((ext_vector_type(2))) float v2f;
typedef __attribute__((ext_vector_type(8))) float v8f;

#define LMBD 0.1f

// Analysis filters (dec_lo = reverse(h), dec_hi = reverse((-1)^k h[L-1-k])).
// Cross-correlation form: a[j] = sum_t g[t] * x[(2j + t - p) mod N], p = L-2.
__constant__ float DECLO[5][10] = {
  { 0.7071067811865476f, 0.7071067811865476f, 0.f,0.f,0.f,0.f,0.f,0.f,0.f,0.f },
  { -0.12940952255126037f, 0.2241438680420134f, 0.8365163037378079f,
    0.48296291314453416f, 0.f,0.f,0.f,0.f,0.f,0.f },
  { 0.03522629188570953f, -0.08544127388202666f, -0.13501102001025458f,
    0.4598775021184914f, 0.8068915093110924f, 0.3326705529500825f, 0.f,0.f,0.f,0.f },
  { -0.010597401785069032f, 0.0328830116668852f, 0.030841381835560763f,
    -0.18703481171909308f, -0.027983769416859854f, 0.6308807679298587f,
    0.7148465705529154f, 0.2303778133088964f, 0.f,0.f },
  { 0.0033357252854737712f, -0.012580751999081999f, -0.006241490212798274f,
    0.07757149384004572f, -0.032244869584638375f, -0.24229488706638203f,
    0.13842814590132074f, 0.7243085284377729f, 0.6038292697971896f,
    0.16010239797419293f }
};
__constant__ float DECHI[5][10] = {
  { -0.7071067811865476f, 0.7071067811865476f, 0.f,0.f,0.f,0.f,0.f,0.f,0.f,0.f },
  { -0.48296291314453416f, 0.8365163037378079f, -0.2241438680420134f,
    -0.12940952255126037f, 0.f,0.f,0.f,0.f,0.f,0.f },
  { -0.3326705529500825f, 0.8068915093110924f, -0.4598775021184914f,
    -0.13501102001025458f, 0.08544127388202666f, 0.03522629188570953f, 0.f,0.f,0.f,0.f },
  { -0.2303778133088964f, 0.7148465705529154f, -0.6308807679298587f,
    -0.027983769416859854f, 0.18703481171909308f, 0.030841381835560763f,
    -0.0328830116668852f, -0.010597401785069032f, 0.f,0.f },
  { -0.16010239797419293f, 0.6038292697971896f, -0.7243085284377729f,
    0.13842814590132074f, 0.24229488706638203f, -0.032244869584638375f,
    -0.07757149384004572f, -0.006241490212798274f, 0.012580751999081999f,
    0.0033357252854737712f }
};

#define SIN_ROWS 104
#define SIN_COLS 112
#define SW 114   // sIn row stride (floats)
#define CW 114   // sCoef row stride
#define DW 50    // shrink-delta stride (48 cols)
#define TW 66    // t3 stride (64 cols)

#define NFT (2*10*32*2)     // FWD band table floats (v2f per filt,k,lane)
#define NIT (2*4*3*32*2)    // INV band table floats (v2f per filt,Q,kk,lane)
#define LDS_FLOATS (SIN_ROWS*SW + 2*48*CW + 4*48*DW + 2*48*TW + NFT + NIT)

__device__ __forceinline__ float bv(const float* g, int L, int t) {
  return (t >= 0 && t < L) ? g[t] : 0.0f;
}

__device__ __forceinline__ v8f wmma4(v2f a, v2f b, v8f c) {
  return __builtin_amdgcn_wmma_f32_16x16x4_f32(false, a, false, b,
                                               (short)0, c, false, false);
}

template<int L>
__device__ __forceinline__ void wavelet_pass(
    float* sIn, float* sCoef, float* sD, float* sT3, float* sFT, float* sIT,
    const float* glo, const float* ghi, float wi,
    int tid, int wave, int lane, int lh, int ll, v8f& accO) {
  constexpr int p  = L - 2;
  constexpr int KF = (29 + L) / 4 + 1;   // forward band K-chunks (of 4)

  // ---- Build band-fragment tables for this wavelet (input-independent).
  // FWD[f][k][lane] = A/B fragment of g[c - 2m] band (c = 4k+2lh+{0,1}, m = ll).
  // INV[f][Q][kk][lane] = fragment of g[X - 2K + p + 8] (X = 16Q+ll,
  //                       K = 4(2Q+1+kk)+2lh+{0,1}); 3 chunks cover the band.
  for (int e = tid; e < 640 + 768; e += 512) {
    int ln = e & 31, lh2 = ln >> 4, ll2 = ln & 15;
    if (e < 640) {
      int f = e / 320, k = (e >> 5) % 10;
      const float* g = f ? ghi : glo;
      int c = 4 * k + 2 * lh2 - 2 * ll2;
      sFT[e * 2]     = bv(g, L, c);
      sFT[e * 2 + 1] = bv(g, L, c + 1);
    } else {
      int e2 = e - 640;
      int f = e2 / 384, r = e2 % 384, Q = r / 96, kk = (r >> 5) % 3;
      const float* g = f ? ghi : glo;
      int K = 4 * (2 * Q + 1 + kk) + 2 * lh2;
      int t0 = (16 * Q + ll2) - 2 * K + p + 8;
      sIT[e2 * 2]     = bv(g, L, t0);
      sIT[e2 * 2 + 1] = bv(g, L, t0 - 2);
    }
  }
  __syncthreads();

  // ---- Stage 1: vertical forward.  coefV[f][48][cols] (coef base r0/2 - 4).
  for (int job = wave; job < 42; job += 16) {
    int f = job / 21, r = job % 21, m0 = r / 7, n0 = r % 7;
    const float* base = sIn + (32 * m0 + (8 - p)) * SW + 16 * n0 + ll;
    v2f a[KF], b[KF];
#pragma unroll
    for (int k = 0; k < KF; ++k) {
      a[k] = *(const v2f*)(sFT + ((f * 10 + k) * 32 + lane) * 2);
      b[k].x = base[(4 * k + 2 * lh) * SW];
      b[k].y = base[(4 * k + 2 * lh + 1) * SW];
    }
    v8f acc = {};
#pragma unroll
    for (int k = 0; k < KF; ++k) acc = wmma4(a[k], b[k], acc);
    float* dst = sCoef + f * 48 * CW + 16 * n0 + ll;
#pragma unroll
    for (int v = 0; v < 8; ++v) dst[(16 * m0 + v + 8 * lh) * CW] = acc[v];
  }
  __syncthreads();

  // ---- Stage 2: horizontal forward, then shrink+weight on the D fragments:
  // d = w_i*(softshrink(a)-a) = -w_i*clamp(a, -λ, λ)  (elementwise).
  for (int job = wave; job < 36; job += 16) {
    int cmb = job / 9, r = job % 9, m0 = r / 3, n0 = r % 3;
    int fv = cmb >> 1, fh = cmb & 1;
    const float* arow = sCoef + fv * 48 * CW + (16 * m0 + ll) * CW
                      + 32 * n0 + (8 - p);
    v2f a[KF], b[KF];
#pragma unroll
    for (int k = 0; k < KF; ++k) {
      a[k] = *(const v2f*)(arow + 4 * k + 2 * lh);     // even offset: b64 load
      b[k] = *(const v2f*)(sFT + ((fh * 10 + k) * 32 + lane) * 2);
    }
    v8f acc = {};
#pragma unroll
    for (int k = 0; k < KF; ++k) acc = wmma4(a[k], b[k], acc);
    float* dst = sD + cmb * 48 * DW + 16 * n0 + ll;
#pragma unroll
    for (int v = 0; v < 8; ++v)
      dst[(16 * m0 + v + 8 * lh) * DW] = -wi * fminf(fmaxf(acc[v], -LMBD), LMBD);
  }
  __syncthreads();

  // ---- Stage 3: horizontal adjoint, summed over h-filter (6 wmmas/job).
  for (int job = wave; job < 24; job += 16) {
    int fv = job / 12, r = job % 12, m0 = r / 4, q = r % 4;
    v2f a[6], b[6];
#pragma unroll
    for (int fh = 0; fh < 2; ++fh)
#pragma unroll
      for (int kk = 0; kk < 3; ++kk) {
        int kc = 4 * (2 * q + 1 + kk) + 2 * lh;
        a[fh * 3 + kk] = *(const v2f*)(sD + (fv * 2 + fh) * 48 * DW
                                        + (16 * m0 + ll) * DW + kc);
        b[fh * 3 + kk] = *(const v2f*)(sIT + (((fh * 4 + q) * 3 + kk) * 32 + lane) * 2);
      }
    v8f acc = {};
#pragma unroll
    for (int i = 0; i < 6; ++i) acc = wmma4(a[i], b[i], acc);
    float* dst = sT3 + fv * 48 * TW + 16 * q + ll;
#pragma unroll
    for (int v = 0; v < 8; ++v) dst[(16 * m0 + v + 8 * lh) * TW] = acc[v];
  }
  __syncthreads();

  // ---- Stage 4: vertical adjoint; each wave owns one 16x16 output tile and
  // accumulates through the WMMA C operand, chained across all wavelets.
  {
    int m0 = wave >> 2, n0 = wave & 3;
    v2f a[6], b[6];
#pragma unroll
    for (int fv = 0; fv < 2; ++fv)
#pragma unroll
      for (int kk = 0; kk < 3; ++kk) {
        int kr = 4 * (2 * m0 + 1 + kk) + 2 * lh;
        a[fv * 3 + kk] = *(const v2f*)(sIT + (((fv * 4 + m0) * 3 + kk) * 32 + lane) * 2);
        b[fv * 3 + kk].x = sT3[fv * 48 * TW + kr * TW + 16 * n0 + ll];
        b[fv * 3 + kk].y = sT3[fv * 48 * TW + (kr + 1) * TW + 16 * n0 + ll];
      }
#pragma unroll
    for (int i = 0; i < 6; ++i) accO = wmma4(a[i], b[i], accO);
  }
  __syncthreads();
}

__global__ __launch_bounds__(512)
void wdenoise_kernel(const float* __restrict__ x, const float* __restrict__ wp,
                     float* __restrict__ out) {
  extern __shared__ float lds[];
  float* sIn   = lds;                        // [104][114] halo window
  float* sCoef = sIn + SIN_ROWS * SW;        // [2][48][114] vertical-fwd coefs
  float* sD    = sCoef + 2 * 48 * CW;        // [4][48][50]  weighted deltas
  float* sT3   = sD + 4 * 48 * DW;           // [2][48][66]  horizontal-adjoint
  float* sFT   = sT3 + 2 * 48 * TW;          // FWD band table
  float* sIT   = sFT + NFT;                  // INV band table

  const int tid  = threadIdx.x;
  const int lane = tid & 31;
  const int wave = tid >> 5;
  const int lh   = lane >> 4;
  const int ll   = lane & 15;

  const int r0 = blockIdx.y * 64;
  const int c0 = blockIdx.x * 64;
  const size_t planeOff = (size_t)blockIdx.z << 20;

  // Load 104x112 halo window with periodic wrap (covers db1..db5 halos).
  for (int i = tid; i < SIN_ROWS * SIN_COLS; i += 512) {
    int r = i / SIN_COLS, c = i % SIN_COLS;
    int gr = (r0 - 16 + r + 1024) & 1023;
    int gc = (c0 - 16 + c + 1024) & 1023;
    sIn[r * SW + c] = x[planeOff + (size_t)gr * 1024 + gc];
  }

  float w5[5]; float wsum = 0.f;
#pragma unroll
  for (int i = 0; i < 5; ++i) { w5[i] = wp[i]; wsum += w5[i]; }
  const float winv = 1.0f / wsum;

  v8f acc = {};
  __syncthreads();

  wavelet_pass<2>(sIn, sCoef, sD, sT3, sFT, sIT, DECLO[0], DECHI[0],
                  w5[0] * winv, tid, wave, lane, lh, ll, acc);
  wavelet_pass<4>(sIn, sCoef, sD, sT3, sFT, sIT, DECLO[1], DECHI[1],
                  w5[1] * winv, tid, wave, lane, lh, ll, acc);
  wavelet_pass<6>(sIn, sCoef, sD, sT3, sFT, sIT, DECLO[2], DECHI[2],
                  w5[2] * winv, tid, wave, lane, lh, ll, acc);
  wavelet_pass<8>(sIn, sCoef, sD, sT3, sFT, sIT, DECLO[3], DECHI[3],
                  w5[3] * winv, tid, wave, lane, lh, ll, acc);
  wavelet_pass<10>(sIn, sCoef, sD, sT3, sFT, sIT, DECLO[4], DECHI[4],
                   w5[4] * winv, tid, wave, lane, lh, ll, acc);

  // Epilogue: out = x + sum_i w_i * delta_i (x re-read from the LDS window).
  {
    int m0 = wave >> 2, n0 = wave & 3;
#pragma unroll
    for (int v = 0; v < 8; ++v) {
      int rl = 16 * m0 + v + 8 * lh;
      int cl = 16 * n0 + ll;
      float xv = sIn[(16 + rl) * SW + (16 + cl)];
      out[planeOff + (size_t)(r0 + rl) * 1024 + (c0 + cl)] = xv + acc[v];
    }
  }
}

extern "C" void kernel_launch(void* const* d_in, const int* in_sizes, int n_in,
                              void* d_out, int out_size, void* d_ws, size_t ws_size,
                              hipStream_t stream) {
  (void)in_sizes; (void)n_in; (void)d_ws; (void)ws_size; (void)out_size;
  const float* x  = (const float*)d_in[0];
  const float* wp = (const float*)d_in[1];
  float* outp = (float*)d_out;

  const size_t shmem = (size_t)LDS_FLOATS * sizeof(float);   // ~166 KB
  hipFuncSetAttribute((const void*)wdenoise_kernel,
                      hipFuncAttributeMaxDynamicSharedMemorySize, (int)shmem);

  dim3 grid(16, 16, 24);   // 16x16 tiles of 64x64 per plane, 24 planes
  dim3 block(512);         // 16 wave32 waves
  hipLaunchKernelGGL(wdenoise_kernel, grid, block, shmem, stream, x, wp, outp);
}